// PreImage_21861383536877
// MI455X (gfx1250) — compile-verified
//
#include <hip/hip_runtime.h>
#include <cstdint>

// ---------------------------------------------------------------------------
// Graph message passing: out[tgt[k]] += x[src[k]] * e[k]  (N=100000, E=1.6M, D=64)
// Atomic/bandwidth bound (AI ~2.5 FLOP/byte) -> no WMMA role. CDNA5 paths:
//   - global_load_async_to_lds_b128 (ASYNCcnt) double-buffered edge staging
//   - partial s_wait_asynccnt 3 (in-order completion) to overlap stage+compute
//   - 128-bit coalesced gathers of x rows
//   - global_atomic_add_f32 (no-return, STOREcnt) scatter
//   - global_prefetch_b8 on next gather row
// ---------------------------------------------------------------------------

#define D_DIM    64
#define TILE     512      // edges per staged tile (6 KB LDS per buffer)
#define THREADS  256      // 8 waves (wave32)

__global__ void zero_out_kernel(float4* __restrict__ out4, int n4) {
    int i = blockIdx.x * blockDim.x + threadIdx.x;
    if (i < n4) out4[i] = make_float4(0.f, 0.f, 0.f, 0.f);
}

// Tail edges (E % TILE) handled directly from global memory (rare/zero path).
__global__ void edge_tail_kernel(const float* __restrict__ x,
                                 const int*   __restrict__ src,
                                 const int*   __restrict__ tgt,
                                 const float* __restrict__ e,
                                 float*       __restrict__ out,
                                 int e0, int n_edges) {
    int k = e0 + blockIdx.x * blockDim.x + threadIdx.x;
    if (k >= n_edges) return;
    const int   s = src[k];
    const int   t = tgt[k];
    const float w = e[k];
    const float4* xp = (const float4*)(x + (size_t)s * D_DIM);
    float*        op = out + (size_t)t * D_DIM;
    #pragma unroll
    for (int d4 = 0; d4 < D_DIM / 4; ++d4) {
        const float4 v = xp[d4];
        unsafeAtomicAdd(op + d4 * 4 + 0, v.x * w);
        unsafeAtomicAdd(op + d4 * 4 + 1, v.y * w);
        unsafeAtomicAdd(op + d4 * 4 + 2, v.z * w);
        unsafeAtomicAdd(op + d4 * 4 + 3, v.w * w);
    }
}

__global__ __launch_bounds__(THREADS)
void edge_scatter_kernel(const float* __restrict__ x,
                         const int*   __restrict__ src,
                         const int*   __restrict__ tgt,
                         const float* __restrict__ e,
                         float*       __restrict__ out,
                         int ntiles) {
    __shared__ int   s_src[2][TILE];
    __shared__ int   s_tgt[2][TILE];
    __shared__ float s_e[2][TILE];

    const int tid = threadIdx.x;

    // --- async stage: 128 threads x 3 x B128 cover one 512-edge tile --------
    auto stage = [&](int buf, int tile0) {
        if (tid < (TILE / 4)) {
            // generic LDS pointer: low 32 bits == LDS byte offset (ISA §10.2)
            unsigned l_src = (unsigned)(uintptr_t)(&s_src[buf][0]) + (unsigned)(tid * 16);
            unsigned l_tgt = (unsigned)(uintptr_t)(&s_tgt[buf][0]) + (unsigned)(tid * 16);
            unsigned l_e   = (unsigned)(uintptr_t)(&s_e[buf][0])   + (unsigned)(tid * 16);
            const void* g_src = (const void*)(src + tile0 + tid * 4);
            const void* g_tgt = (const void*)(tgt + tile0 + tid * 4);
            const void* g_e   = (const void*)(e   + tile0 + tid * 4);
            asm volatile("global_load_async_to_lds_b128 %0, %1, off"
                         :: "v"(l_src), "v"(g_src) : "memory");
            asm volatile("global_load_async_to_lds_b128 %0, %1, off"
                         :: "v"(l_tgt), "v"(g_tgt) : "memory");
            asm volatile("global_load_async_to_lds_b128 %0, %1, off"
                         :: "v"(l_e),   "v"(g_e)   : "memory");
        }
    };

    // --- process one staged tile: 2 edges per wave iteration ---------------
    const int wave = tid >> 5;          // 0..7
    const int lane = tid & 31;
    const int sub  = lane >> 4;         // which of the 2 edges this half-wave owns
    const int l16  = lane & 15;         // lane within the edge's 16-lane group

    auto process = [&](int buf, int tile0) {
        (void)tile0;
        for (int el2 = wave * 2; el2 < TILE; el2 += 16) {
            const int el = el2 + sub;
            const int   s = s_src[buf][el];   // LDS broadcast reads
            const int   t = s_tgt[buf][el];
            const float w = s_e[buf][el];

            if (el + 16 < TILE) {             // prefetch next edge's x row
                const int sn = s_src[buf][el + 16];
                __builtin_prefetch(x + (size_t)sn * D_DIM, 0, 0); // global_prefetch_b8
            }

            const float4 v =
                ((const float4*)(x + (size_t)s * D_DIM))[l16];    // global_load_b128

            float* op = out + (size_t)t * D_DIM + l16 * 4;
            unsafeAtomicAdd(op + 0, v.x * w);   // global_atomic_add_f32 (no-return)
            unsafeAtomicAdd(op + 1, v.y * w);
            unsafeAtomicAdd(op + 2, v.z * w);
            unsafeAtomicAdd(op + 3, v.w * w);
        }
    };

    // --- persistent double-buffered pipeline over tiles --------------------
    int t = blockIdx.x;
    if (t >= ntiles) return;
    int cur = 0;
    stage(cur, t * TILE);

    for (;;) {
        const int  next     = t + gridDim.x;
        const bool has_next = next < ntiles;
        if (has_next) {
            stage(1 - cur, next * TILE);
            // in-order ASYNCcnt: <=3 outstanding means current buffer's 3
            // loads are done while next buffer's 3 remain in flight
            asm volatile("s_wait_asynccnt 0x3" ::: "memory");
        } else {
            asm volatile("s_wait_asynccnt 0x0" ::: "memory");
        }
        __syncthreads();                 // publish staged LDS to all waves

        process(cur, t * TILE);

        if (!has_next) break;
        __syncthreads();                 // everyone done reading buf[cur]
                                         // before it is re-staged next round
        cur ^= 1;
        t = next;
    }
}

extern "C" void kernel_launch(void* const* d_in, const int* in_sizes, int n_in,
                              void* d_out, int out_size, void* d_ws, size_t ws_size,
                              hipStream_t stream) {
    (void)n_in; (void)d_ws; (void)ws_size;

    const float* x = (const float*)d_in[0];   // [N*D] row-major
    const int*   a = (const int*)d_in[1];     // [2*E]: src then tgt
    const float* e = (const float*)d_in[2];   // [E]

    float* out = (float*)d_out;               // [N*D]
    const int n_edges = in_sizes[2];          // E
    const int* src = a;
    const int* tgt = a + n_edges;

    // out is accumulated with atomics -> must be zeroed every call
    const int n4 = out_size / 4;
    zero_out_kernel<<<(n4 + THREADS - 1) / THREADS, THREADS, 0, stream>>>(
        (float4*)out, n4);

    const int ntiles = n_edges / TILE;        // full tiles
    if (ntiles > 0) {
        int nblocks = ntiles < 1024 ? ntiles : 1024;   // persistent blocks
        edge_scatter_kernel<<<nblocks, THREADS, 0, stream>>>(
            x, src, tgt, e, out, ntiles);
    }

    const int e0  = ntiles * TILE;            // tail edges (0 for E=1.6M)
    const int rem = n_edges - e0;
    if (rem > 0) {
        edge_tail_kernel<<<(rem + THREADS - 1) / THREADS, THREADS, 0, stream>>>(
            x, src, tgt, e, out, e0, n_edges);
    }
}